// TopKPooling_15006615733141
// MI455X (gfx1250) — compile-verified
//
#include <hip/hip_runtime.h>
#include <stdint.h>

typedef __attribute__((ext_vector_type(2))) float v2f;
typedef __attribute__((ext_vector_type(4))) float v4f;
typedef __attribute__((ext_vector_type(8))) float v8f;

#define NN   200000
#define CC   256
#define KK   100000
#define NPAD 262144   // 2^18 >= NN

// ---------------------------------------------------------------------------
// 1) scores[i] = tanh(dot(x[i], w) + b)   via V_WMMA_F32_16X16X4_F32
//    One wave -> 16 rows. A: 16x4 f32 tile of x; B: w chunk broadcast across
//    all 16 columns, so every column of D holds the per-row dot product.
// ---------------------------------------------------------------------------
__global__ void score_wmma_kernel(const float* __restrict__ x,
                                  const float* __restrict__ w,
                                  const float* __restrict__ bias,
                                  float* __restrict__ scores, int nRows) {
    const int lane   = threadIdx.x & 31;
    const int waveId = (blockIdx.x * blockDim.x + threadIdx.x) >> 5;
    const int row0   = waveId * 16;
    if (row0 >= nRows) return;

    const int mLane = lane & 15;      // M index 0..15
    const int hi    = lane >> 4;      // 0: K={0,1}, 1: K={2,3} within a chunk
    const float* __restrict__ rowPtr = x + (size_t)(row0 + mLane) * CC;

    v8f acc = {0.f, 0.f, 0.f, 0.f, 0.f, 0.f, 0.f, 0.f};
    #pragma unroll 4
    for (int k0 = 0; k0 < CC; k0 += 4) {
        // A: lane holds x[row, k0+2*hi .. +1]   (16x4 f32 A-layout)
        v2f a = *(const v2f*)(rowPtr + k0 + 2 * hi);
        // B: VGPR v, K = v + 2*hi, value broadcast across all 16 columns
        v2f bb;
        bb.x = w[k0 + 2 * hi];
        bb.y = w[k0 + 2 * hi + 1];
        acc = __builtin_amdgcn_wmma_f32_16x16x4_f32(
            false, a, false, bb, (short)0, acc, false, false);
    }
    // C/D layout: VGPR r -> M=r (lanes 0-15), M=r+8 (lanes 16-31); all N equal
    if (mLane == 0) {
        const float bv = bias[0];
        #pragma unroll
        for (int r = 0; r < 8; ++r) {
            int row = row0 + r + 8 * hi;
            if (row < nRows) scores[row] = tanhf(acc[r] + bv);
        }
    }
}

// ---------------------------------------------------------------------------
// 2) 64-bit sort keys: ascending sort == descending score, ascending index
// ---------------------------------------------------------------------------
__global__ void build_keys_kernel(const float* __restrict__ scores,
                                  unsigned long long* __restrict__ keys) {
    int i = blockIdx.x * blockDim.x + threadIdx.x;
    if (i >= NPAD) return;
    if (i < NN) {
        unsigned u = __float_as_uint(scores[i]);
        unsigned s = (u & 0x80000000u) ? ~u : (u | 0x80000000u); // asc in score
        keys[i] = ((unsigned long long)(~s) << 32) | (unsigned)i;
    } else {
        keys[i] = 0xFFFFFFFFFFFFFFFFull; // padding sorts last
    }
}

// ---------------------------------------------------------------------------
// 3) bitonic compare-exchange pass (2 MB working set -> L2 resident)
// ---------------------------------------------------------------------------
__global__ void bitonic_step_kernel(unsigned long long* __restrict__ keys,
                                    unsigned j, unsigned k) {
    unsigned i   = blockIdx.x * blockDim.x + threadIdx.x;
    unsigned ixj = i ^ j;
    if (ixj > i) {
        unsigned long long a = keys[i];
        unsigned long long b = keys[ixj];
        bool asc = ((i & k) == 0);
        if ((a > b) == asc) { keys[i] = b; keys[ixj] = a; }
    }
}

// ---------------------------------------------------------------------------
// 4) perm / batch_pool / scores_pool outputs + scatter new_mask & sel flags
// ---------------------------------------------------------------------------
__global__ void pool_meta_kernel(const unsigned long long* __restrict__ keys,
                                 const float* __restrict__ scores,
                                 const int* __restrict__ batch,
                                 int* __restrict__ perm_out,
                                 int* __restrict__ batch_out,
                                 float* __restrict__ scores_out,
                                 int* __restrict__ new_mask,
                                 unsigned char* __restrict__ sel) {
    int j = blockIdx.x * blockDim.x + threadIdx.x;
    if (j >= KK) return;
    int idx       = (int)(keys[j] & 0xFFFFFFFFull);
    perm_out[j]   = idx;
    batch_out[j]  = batch[idx];
    scores_out[j] = scores[idx];
    new_mask[idx] = j;
    sel[idx]      = 1;
}

// ---------------------------------------------------------------------------
// 5) x_pool = x[perm]  (streaming: non-temporal b128 both directions)
// ---------------------------------------------------------------------------
__global__ void gather_x_kernel(const float* __restrict__ x,
                                const unsigned long long* __restrict__ keys,
                                float* __restrict__ xpool) {
    int t = blockIdx.x * blockDim.x + threadIdx.x;   // KK * 64 threads
    int j = t >> 6;
    if (j >= KK) return;
    int c  = (t & 63) << 2;
    int src = (int)(keys[j] & 0xFFFFFFFFull);
    const v4f* sp = (const v4f*)(x + (size_t)src * CC + c);
    v4f v = __builtin_nontemporal_load(sp);
    __builtin_nontemporal_store(v, (v4f*)(xpool + (size_t)j * CC + c));
}

// ---------------------------------------------------------------------------
// 6a) per-block (1024 elems) count of non-selected nodes
// ---------------------------------------------------------------------------
__global__ void count_blocks_kernel(const unsigned char* __restrict__ sel,
                                    int* __restrict__ blockCounts) {
    __shared__ int lds[256];
    int base = blockIdx.x * 1024;
    int cnt = 0;
    #pragma unroll
    for (int u = 0; u < 4; ++u) {
        int i = base + threadIdx.x * 4 + u;
        if (i < NN && sel[i] == 0) cnt++;
    }
    lds[threadIdx.x] = cnt;
    __syncthreads();
    for (int s = 128; s > 0; s >>= 1) {
        if (threadIdx.x < (unsigned)s) lds[threadIdx.x] += lds[threadIdx.x + s];
        __syncthreads();
    }
    if (threadIdx.x == 0) blockCounts[blockIdx.x] = lds[0];
}

// ---------------------------------------------------------------------------
// 6b) exclusive scan over block counts (single workgroup, <=256 blocks)
// ---------------------------------------------------------------------------
__global__ void scan_blocks_kernel(const int* __restrict__ counts,
                                   int* __restrict__ offsets, int nB) {
    __shared__ int lds[256];
    int v = (threadIdx.x < (unsigned)nB) ? counts[threadIdx.x] : 0;
    lds[threadIdx.x] = v;
    __syncthreads();
    for (int s = 1; s < 256; s <<= 1) {
        int t = 0;
        if (threadIdx.x >= (unsigned)s) t = lds[threadIdx.x - s];
        __syncthreads();
        lds[threadIdx.x] += t;
        __syncthreads();
    }
    if (threadIdx.x < (unsigned)nB) offsets[threadIdx.x] = lds[threadIdx.x] - v;
}

// ---------------------------------------------------------------------------
// 6c) stable scatter of kept edge slots, fused with edge relabel gather
// ---------------------------------------------------------------------------
__global__ void scatter_edges_kernel(const unsigned char* __restrict__ sel,
                                     const int* __restrict__ offsets,
                                     const int* __restrict__ e0,
                                     const int* __restrict__ e1,
                                     const int* __restrict__ new_mask,
                                     int* __restrict__ eiOut,
                                     int* __restrict__ ejOut) {
    __shared__ int lds[256];
    int base = blockIdx.x * 1024;
    int flags[4];
    int cnt = 0;
    #pragma unroll
    for (int u = 0; u < 4; ++u) {
        int i = base + threadIdx.x * 4 + u;
        flags[u] = (i < NN && sel[i] == 0) ? 1 : 0;
        cnt += flags[u];
    }
    lds[threadIdx.x] = cnt;
    __syncthreads();
    for (int s = 1; s < 256; s <<= 1) {
        int t = 0;
        if (threadIdx.x >= (unsigned)s) t = lds[threadIdx.x - s];
        __syncthreads();
        lds[threadIdx.x] += t;
        __syncthreads();
    }
    int pos = offsets[blockIdx.x] + (lds[threadIdx.x] - cnt); // block off + excl
    #pragma unroll
    for (int u = 0; u < 4; ++u) {
        if (flags[u]) {
            int i = base + threadIdx.x * 4 + u;
            eiOut[pos] = new_mask[e0[i]];
            ejOut[pos] = new_mask[e1[i]];
            pos++;
        }
    }
}

// ---------------------------------------------------------------------------
extern "C" void kernel_launch(void* const* d_in, const int* in_sizes, int n_in,
                              void* d_out, int out_size, void* d_ws, size_t ws_size,
                              hipStream_t stream) {
    const float* x     = (const float*)d_in[0];
    const int*   eidx  = (const int*)d_in[1];
    const int*   batch = (const int*)d_in[2];
    const float* w     = (const float*)d_in[3];
    const float* bias  = (const float*)d_in[4];
    const int* e0 = eidx;
    const int* e1 = eidx + NN;

    // ---- output layout (flat, return order) ----
    float* out_f = (float*)d_out;
    float* xpool      = out_f;                                   // KK*CC
    int*   ei_out     = (int*)(out_f + (size_t)KK * CC);         // NN-KK
    int*   ej_out     = ei_out + (NN - KK);                      // NN-KK
    int*   perm_out   = ej_out + (NN - KK);                      // KK
    int*   batch_out  = perm_out + KK;                           // KK
    float* scores_out = (float*)(batch_out + KK);                // KK

    // ---- workspace layout ----
    char* ws = (char*)d_ws;
    float*              scores      = (float*)ws;                         // NN*4
    unsigned long long* keys        = (unsigned long long*)(ws + 800000); // NPAD*8
    int*                new_mask    = (int*)(ws + 800000 + (size_t)NPAD * 8);
    unsigned char*      sel         = (unsigned char*)((char*)new_mask + (size_t)NN * 4);
    int*                blockCounts = (int*)((char*)sel + 200704);
    int*                blockOffs   = blockCounts + 256;

    const int nBlocks1024 = (NN + 1023) / 1024;  // 196

    // 1) scores via WMMA matvec: 12500 waves, 8 waves/block
    {
        int waves  = (NN + 15) / 16;
        int blocks = (waves + 7) / 8;
        score_wmma_kernel<<<blocks, 256, 0, stream>>>(x, w, bias, scores, NN);
    }
    // 2) sort keys (padded to 2^18)
    build_keys_kernel<<<NPAD / 256, 256, 0, stream>>>(scores, keys);
    // 3) bitonic sort, fully L2-resident (2 MB)
    for (unsigned k = 2; k <= (unsigned)NPAD; k <<= 1)
        for (unsigned j = k >> 1; j > 0; j >>= 1)
            bitonic_step_kernel<<<NPAD / 256, 256, 0, stream>>>(keys, j, k);
    // 4) zero relabel map + selection flags, then scatter meta outputs
    (void)hipMemsetAsync(new_mask, 0, (size_t)NN * 4, stream);
    (void)hipMemsetAsync(sel, 0, 200704, stream);
    pool_meta_kernel<<<(KK + 255) / 256, 256, 0, stream>>>(
        keys, scores, batch, perm_out, batch_out, scores_out, new_mask, sel);
    // 5) x_pool gather (KK*64 threads, float4 each)
    gather_x_kernel<<<(KK * 64) / 256, 256, 0, stream>>>(x, keys, xpool);
    // 6) stable compaction of non-selected positions + edge relabel
    count_blocks_kernel<<<nBlocks1024, 256, 0, stream>>>(sel, blockCounts);
    scan_blocks_kernel<<<1, 256, 0, stream>>>(blockCounts, blockOffs, nBlocks1024);
    scatter_edges_kernel<<<nBlocks1024, 256, 0, stream>>>(
        sel, blockOffs, e0, e1, new_mask, ei_out, ej_out);
}